// DrugDiseaseHGT_8942121910408
// MI455X (gfx1250) — compile-verified
//
#include <hip/hip_runtime.h>
#include <math.h>

// ---------------------------------------------------------------------------
// DrugDiseaseHGT forward for MI455X (gfx1250).
// fp32 everywhere (matches reference numerics); dense GEMMs use
// V_WMMA_F32_16X16X4_F32 via LDS-staged operand tiles. Global->LDS staging
// uses gfx1250 async-to-LDS ops when the compiler declares the builtins.
// ---------------------------------------------------------------------------

typedef __attribute__((ext_vector_type(2))) float v2f;
typedef __attribute__((ext_vector_type(8))) float v8f;

#define H_HEADS 4
#define D_DIM   32
#define HID     128

#if defined(__has_builtin)
#if __has_builtin(__builtin_amdgcn_global_load_async_to_lds_b128) && \
    __has_builtin(__builtin_amdgcn_s_wait_asynccnt)
#define USE_ASYNC_LDS 1
#endif
#endif
#ifndef USE_ASYNC_LDS
#define USE_ASYNC_LDS 0
#endif

__device__ __forceinline__ float gelu_exact(float x) {
    return 0.5f * x * (1.0f + erff(x * 0.70710678118654752f));
}

// Monotone float<->uint encoding for atomicMax on floats.
__device__ __forceinline__ unsigned enc_f32(float x) {
    unsigned u = __float_as_uint(x);
    return (u & 0x80000000u) ? ~u : (u | 0x80000000u);
}
__device__ __forceinline__ float dec_f32(unsigned u) {
    u = (u & 0x80000000u) ? (u & 0x7FFFFFFFu) : ~u;
    return __uint_as_float(u);
}

// Copy 16 aligned bytes global -> LDS (async-to-LDS when available).
// Builtin signature (from clang diagnostic): param0 = AS1 pointer to
// 16-byte int vector (global source), param1 = AS3 pointer (LDS dest).
typedef int b128_t __attribute__((vector_size(4 * sizeof(int))));

__device__ __forceinline__ void cp16(const float* __restrict__ g,
                                     float* __restrict__ l) {
#if USE_ASYNC_LDS
    __builtin_amdgcn_global_load_async_to_lds_b128(
        (__attribute__((address_space(1))) b128_t*)g,
        (__attribute__((address_space(3))) b128_t*)l, 0, 0);
#else
    *(float4*)l = *(const float4*)g;
#endif
}

__device__ __forceinline__ void cp_wait() {
#if USE_ASYNC_LDS
    __builtin_amdgcn_s_wait_asynccnt(0);
#endif
}

// ---------------------------------------------------------------------------
// LDS-staged fp32 WMMA GEMM: C[M,N] = A[M,K] @ B[K,N] + bias.
// Block = 128 threads (4 waves). Block computes a 16x64 C panel:
//   sA = A tile 16xK (row stride K+4), sB = B panel Kx64 (row stride 68),
//   wave w computes the 16x16 sub-tile at panel columns w*16..w*16+15.
// Requires M%16==0, N%64==0, K%4==0 (true for every call site here).
// ---------------------------------------------------------------------------
__global__ void wmma_gemm_tiled(const float* __restrict__ A, int lda,
                                const float* __restrict__ B, int ldb,
                                const float* __restrict__ bias,
                                float* __restrict__ C, int ldc,
                                int M, int N, int K) {
    extern __shared__ float smem[];
    const int sa_ld = K + 4;
    const int sb_ld = 68;
    float* sA = smem;                 // [16][K+4]
    float* sB = smem + 16 * sa_ld;    // [K][68]

    const int tid  = threadIdx.x;     // 0..127
    const int row0 = blockIdx.x * 16;
    const int colp = blockIdx.y * 64;

    // stage A tile: 16 x K, 4 floats per transfer
    const int kq = K >> 2;            // transfers per row
    for (int i = tid; i < 16 * kq; i += 128) {
        const int r  = i / kq;
        const int k4 = (i - r * kq) << 2;
        cp16(A + (size_t)(row0 + r) * lda + k4, sA + r * sa_ld + k4);
    }
    // stage B panel: K x 64, 4 floats per transfer
    for (int j = tid; j < K * 16; j += 128) {
        const int r  = j >> 4;
        const int c4 = (j & 15) << 2;
        cp16(B + (size_t)r * ldb + colp + c4, sB + r * sb_ld + c4);
    }
    cp_wait();
    __syncthreads();

    const int wave = tid >> 5, lane = tid & 31;
    const int half = lane >> 4, l16 = lane & 15;
    const int wcol = wave * 16;           // column offset inside panel
    const int col  = colp + wcol + l16;   // global output column

    v8f acc = {0.f, 0.f, 0.f, 0.f, 0.f, 0.f, 0.f, 0.f};
    for (int k0 = 0; k0 < K; k0 += 4) {
        const int ka = k0 + 2 * half;
        v2f af; af.x = sA[l16 * sa_ld + ka];
                af.y = sA[l16 * sa_ld + ka + 1];
        v2f bf; bf.x = sB[ka * sb_ld + wcol + l16];
                bf.y = sB[(ka + 1) * sb_ld + wcol + l16];
        acc = __builtin_amdgcn_wmma_f32_16x16x4_f32(
            false, af, false, bf, (short)0, acc, false, false);
    }

    const float bv = bias[col];
#pragma unroll
    for (int r = 0; r < 8; ++r)
        C[(size_t)(row0 + r + half * 8) * ldc + col] = acc[r] + bv;
}

// ---------------------------------------------------------------------------
// Elementwise exact GELU in place.
// ---------------------------------------------------------------------------
__global__ void gelu_inplace_k(float* __restrict__ p, size_t n) {
    const size_t i = blockIdx.x * (size_t)blockDim.x + threadIdx.x;
    if (i < n) p[i] = gelu_exact(p[i]);
}

// ---------------------------------------------------------------------------
// Pass 1 per edge type: logits + segment max.
// One wave per edge. logit[e,h] = (k_src[h,:] . (a_rel[h] @ q_dst[h,:])) *
//                                 p_rel[h] / sqrt(D)
// ---------------------------------------------------------------------------
__global__ void edge_logit_k(const int* __restrict__ ei, int E_,
                             const float* __restrict__ Kb,
                             const float* __restrict__ Qb,
                             int src_off, int dst_off,
                             const float* __restrict__ a_rel,  // [H,32,32]
                             const float* __restrict__ p_rel,  // [H]
                             float* __restrict__ logit,        // [E,H]
                             unsigned* __restrict__ smax) {    // [Ntot,H]
    __shared__ float sar[H_HEADS * D_DIM * D_DIM];  // 16 KB
    for (int i = threadIdx.x; i < H_HEADS * D_DIM * D_DIM; i += blockDim.x)
        sar[i] = a_rel[i];
    __syncthreads();

    const int wave = (int)((blockIdx.x * (size_t)blockDim.x + threadIdx.x) >> 5);
    const int lane = threadIdx.x & 31;
    if (wave >= E_) return;
    const int src = ei[wave] + src_off;
    const int dst = ei[E_ + wave] + dst_off;
    const float scale = 0.17677669529663687f;  // 1/sqrt(32)

    for (int h = 0; h < H_HEADS; ++h) {
        const float kv = Kb[(size_t)src * HID + h * D_DIM + lane];
        const float qv = Qb[(size_t)dst * HID + h * D_DIM + lane];
        const float* ar = sar + h * (D_DIM * D_DIM) + lane * D_DIM;
        float inner = 0.0f;
#pragma unroll
        for (int f = 0; f < D_DIM; ++f)
            inner += ar[f] * __shfl(qv, f, 32);
        float part = kv * inner;  // lane indexes d
#pragma unroll
        for (int off = 16; off > 0; off >>= 1)
            part += __shfl_xor(part, off, 32);
        if (lane == 0) {
            const float L = part * p_rel[h] * scale;
            logit[(size_t)wave * H_HEADS + h] = L;
            atomicMax(&smax[(size_t)dst * H_HEADS + h], enc_f32(L));
        }
    }
}

// ---------------------------------------------------------------------------
// Pass 2: e = exp(logit - m); segment sum. One thread per (edge,head).
// ---------------------------------------------------------------------------
__global__ void edge_exp_k(const int* __restrict__ ei, int E_, int dst_off,
                           float* __restrict__ logit,
                           const unsigned* __restrict__ smax,
                           float* __restrict__ ssum) {
    const int idx = blockIdx.x * blockDim.x + threadIdx.x;
    if (idx >= E_ * H_HEADS) return;
    const int e = idx >> 2;
    const int h = idx & 3;
    const int dst = ei[E_ + e] + dst_off;
    float m = dec_f32(smax[(size_t)dst * H_HEADS + h]);
    if (m < -3.0e38f) m = 0.0f;  // empty-segment guard (mirrors reference)
    const float ex = expf(logit[idx] - m);
    logit[idx] = ex;
    atomicAdd(&ssum[(size_t)dst * H_HEADS + h], ex);
}

// ---------------------------------------------------------------------------
// Pass 3: alpha-weighted message v_e = v_src @ m_rel, scatter-add into agg.
// One wave per edge.
// ---------------------------------------------------------------------------
__global__ void edge_msg_k(const int* __restrict__ ei, int E_,
                           const float* __restrict__ Vb,
                           int src_off, int dst_off,
                           const float* __restrict__ m_rel,   // [H,32,32]
                           const float* __restrict__ logit,   // holds exp()
                           const float* __restrict__ ssum,
                           float* __restrict__ agg) {
    __shared__ float smr[H_HEADS * D_DIM * D_DIM];  // 16 KB
    for (int i = threadIdx.x; i < H_HEADS * D_DIM * D_DIM; i += blockDim.x)
        smr[i] = m_rel[i];
    __syncthreads();

    const int wave = (int)((blockIdx.x * (size_t)blockDim.x + threadIdx.x) >> 5);
    const int lane = threadIdx.x & 31;
    if (wave >= E_) return;
    const int src = ei[wave] + src_off;
    const int dst = ei[E_ + wave] + dst_off;

    for (int h = 0; h < H_HEADS; ++h) {
        const float alpha = logit[(size_t)wave * H_HEADS + h] /
                            (ssum[(size_t)dst * H_HEADS + h] + 1e-16f);
        const float vv = Vb[(size_t)src * HID + h * D_DIM + lane];
        float ve = 0.0f;  // lane indexes f
#pragma unroll
        for (int d = 0; d < D_DIM; ++d)
            ve += __shfl(vv, d, 32) * smr[h * (D_DIM * D_DIM) + d * D_DIM + lane];
        atomicAdd(&agg[(size_t)dst * HID + h * D_DIM + lane], alpha * ve);
    }
}

// ---------------------------------------------------------------------------
// Residual skip-gate + LayerNorm + GELU. One wave per node (128 cols).
// x_new = gelu(LN(beta*o + (1-beta)*x + x) * g + b)
// ---------------------------------------------------------------------------
__global__ void ln_update_k(const float* __restrict__ CO, float* __restrict__ X,
                            const float* __restrict__ skip,
                            const float* __restrict__ g,
                            const float* __restrict__ b,
                            int node_off, int Nnt) {
    const int wave = (int)((blockIdx.x * (size_t)blockDim.x + threadIdx.x) >> 5);
    const int lane = threadIdx.x & 31;
    if (wave >= Nnt) return;
    const size_t base = (size_t)(node_off + wave) * HID;
    const float beta = 1.0f / (1.0f + expf(-skip[0]));

    float t[4];
    float s = 0.0f, s2 = 0.0f;
#pragma unroll
    for (int j = 0; j < 4; ++j) {
        const int c = lane * 4 + j;
        const float o = CO[base + c];
        const float x = X[base + c];
        const float v = beta * o + (1.0f - beta) * x + x;
        t[j] = v; s += v; s2 += v * v;
    }
#pragma unroll
    for (int off = 16; off > 0; off >>= 1) {
        s  += __shfl_xor(s,  off, 32);
        s2 += __shfl_xor(s2, off, 32);
    }
    const float mean = s * (1.0f / 128.0f);
    const float var  = s2 * (1.0f / 128.0f) - mean * mean;
    const float rstd = rsqrtf(var + 1e-5f);
#pragma unroll
    for (int j = 0; j < 4; ++j) {
        const int c = lane * 4 + j;
        const float y = (t[j] - mean) * rstd * g[c] + b[c];
        X[base + c] = gelu_exact(y);
    }
}

// ---------------------------------------------------------------------------
// Fused decoder: 16 pairs / block (128 threads = 4 waves).
// pair = xd[di]*xs[si]; h1 = gelu(pair@W1+b1) [16x256]; h2 = gelu(h1@W2+b2)
// [16x128]; out = h2@W3 + b3. Staged through LDS; GEMMs via fp32 WMMA.
// ---------------------------------------------------------------------------
__global__ void decoder_fused(const float* __restrict__ Xd,
                              const float* __restrict__ Xs,
                              const int* __restrict__ di,
                              const int* __restrict__ si,
                              const float* __restrict__ W1, const float* __restrict__ b1,
                              const float* __restrict__ W2, const float* __restrict__ b2,
                              const float* __restrict__ W3, const float* __restrict__ b3,
                              float* __restrict__ out, int npairs) {
    __shared__ float sp[16 * 128];
    __shared__ float sh1[16 * 256];
    __shared__ float sh2[16 * 128];

    const int p0  = blockIdx.x * 16;   // npairs % 16 == 0 at the call site
    const int tid = threadIdx.x;

    for (int idx = tid; idx < 16 * 128; idx += 128) {
        const int r = idx >> 7, c = idx & 127;
        const int pp = p0 + r;
        sp[idx] = Xd[(size_t)di[pp] * HID + c] * Xs[(size_t)si[pp] * HID + c];
    }
    __syncthreads();

    const int wave = tid >> 5, lane = tid & 31;
    const int half = lane >> 4, l16 = lane & 15;

    // h1: 16x256, K=128 -> 16 col-tiles across 4 waves
    for (int t = wave; t < 16; t += 4) {
        const int col = t * 16 + l16;
        v8f acc = {0.f, 0.f, 0.f, 0.f, 0.f, 0.f, 0.f, 0.f};
        for (int k0 = 0; k0 < 128; k0 += 4) {
            const int ka = k0 + 2 * half;
            v2f af; af.x = sp[l16 * 128 + ka]; af.y = sp[l16 * 128 + ka + 1];
            v2f bf; bf.x = W1[(size_t)ka * 256 + col];
                    bf.y = W1[(size_t)(ka + 1) * 256 + col];
            acc = __builtin_amdgcn_wmma_f32_16x16x4_f32(
                false, af, false, bf, (short)0, acc, false, false);
        }
        const float bv = b1[col];
#pragma unroll
        for (int r = 0; r < 8; ++r)
            sh1[(r + half * 8) * 256 + col] = gelu_exact(acc[r] + bv);
    }
    __syncthreads();

    // h2: 16x128, K=256 -> 8 col-tiles across 4 waves
    for (int t = wave; t < 8; t += 4) {
        const int col = t * 16 + l16;
        v8f acc = {0.f, 0.f, 0.f, 0.f, 0.f, 0.f, 0.f, 0.f};
        for (int k0 = 0; k0 < 256; k0 += 4) {
            const int ka = k0 + 2 * half;
            v2f af; af.x = sh1[l16 * 256 + ka]; af.y = sh1[l16 * 256 + ka + 1];
            v2f bf; bf.x = W2[(size_t)ka * 128 + col];
                    bf.y = W2[(size_t)(ka + 1) * 128 + col];
            acc = __builtin_amdgcn_wmma_f32_16x16x4_f32(
                false, af, false, bf, (short)0, acc, false, false);
        }
        const float bv = b2[col];
#pragma unroll
        for (int r = 0; r < 8; ++r)
            sh2[(r + half * 8) * 128 + col] = gelu_exact(acc[r] + bv);
    }
    __syncthreads();

    if (tid < 16) {
        float s = 0.0f;
        for (int k = 0; k < 128; ++k) s += sh2[tid * 128 + k] * W3[k];
        out[p0 + tid] = s + b3[0];
    }
}

__global__ void fill_u32_k(unsigned* __restrict__ p, unsigned v, size_t n) {
    const size_t i = blockIdx.x * (size_t)blockDim.x + threadIdx.x;
    if (i < n) p[i] = v;
}

// ---------------------------------------------------------------------------
// Host orchestration
// ---------------------------------------------------------------------------
extern "C" void kernel_launch(void* const* d_in, const int* in_sizes, int n_in,
                              void* d_out, int out_size, void* d_ws, size_t ws_size,
                              hipStream_t stream) {
    (void)in_sizes; (void)n_in; (void)out_size; (void)ws_size;

    const int NN[3]   = {20000, 8000, 30000};
    const int OFF[3]  = {0, 20000, 28000};
    const int NTOT    = 58000;
    const int EE      = 150000;
    const int NPAIRS  = 200000;
    const int ESRC[6] = {0, 1, 0, 2, 2, 1};  // drug,disease,gene = 0,1,2
    const int EDST[6] = {1, 0, 2, 0, 1, 2};

    // ---- inputs (insertion order of setup_inputs, depth-first) ----
    const float* xin[3] = {(const float*)d_in[0], (const float*)d_in[1],
                           (const float*)d_in[2]};
    const int* ei[6];
    for (int i = 0; i < 6; ++i) ei[i] = (const int*)d_in[3 + i];
    const int* drug_idx = (const int*)d_in[9];
    const int* dis_idx  = (const int*)d_in[10];

    int cur = 11;
    auto FP = [&](void) { return (const float*)d_in[cur++]; };

    const float *projW[3], *projB[3];
    for (int t = 0; t < 3; ++t) { projW[t] = FP(); projB[t] = FP(); }

    struct LayerP {
        const float *kW[3], *kB[3], *qW[3], *qB[3], *vW[3], *vB[3], *aW[3], *aB[3];
        const float *skip[3], *lng[3], *lnb[3];
        const float *arel[6], *mrel[6], *prel[6];
    } L[2];
    for (int l = 0; l < 2; ++l) {
        for (int t = 0; t < 3; ++t) { L[l].kW[t] = FP(); L[l].kB[t] = FP(); }
        for (int t = 0; t < 3; ++t) { L[l].qW[t] = FP(); L[l].qB[t] = FP(); }
        for (int t = 0; t < 3; ++t) { L[l].vW[t] = FP(); L[l].vB[t] = FP(); }
        for (int t = 0; t < 3; ++t) { L[l].aW[t] = FP(); L[l].aB[t] = FP(); }
        for (int t = 0; t < 3; ++t) { L[l].skip[t] = FP(); }
        for (int t = 0; t < 3; ++t) { L[l].lng[t] = FP(); L[l].lnb[t] = FP(); }
        for (int e = 0; e < 6; ++e) { L[l].arel[e] = FP(); }
        for (int e = 0; e < 6; ++e) { L[l].mrel[e] = FP(); }
        for (int e = 0; e < 6; ++e) { L[l].prel[e] = FP(); }
    }
    const float *dW[3], *dB[3];
    for (int i = 0; i < 3; ++i) { dW[i] = FP(); dB[i] = FP(); }

    // ---- workspace layout ----
    const size_t F128 = (size_t)NTOT * HID;
    float* ws      = (float*)d_ws;
    float* X       = ws;                 // current features [NTOT,128]
    float* Kb      = X + F128;
    float* Qb      = Kb + F128;
    float* Vb      = Qb + F128;
    float* AGG     = Vb + F128;
    float* LOGIT   = AGG + F128;         // 6*EE*4 floats
    unsigned* SMAX = (unsigned*)(LOGIT + (size_t)6 * EE * H_HEADS);
    float* SSUM    = (float*)(SMAX + (size_t)NTOT * H_HEADS);
    float* CO      = Kb;                 // alias: K dead after edge passes

    const dim3 blk256(256);

    auto launch_gemm = [&](const float* A, int lda, const float* B, int ldb,
                           const float* bias, float* C, int ldc,
                           int M, int N, int K) {
        dim3 grid(M / 16, N / 64);
        const size_t sh = (size_t)(16 * (K + 4) + K * 68) * sizeof(float);
        wmma_gemm_tiled<<<grid, dim3(128), sh, stream>>>(
            A, lda, B, ldb, bias, C, ldc, M, N, K);
    };

    // ---- input projection: X = xin @ Wproj + b (K=256) ----
    for (int t = 0; t < 3; ++t)
        launch_gemm(xin[t], 256, projW[t], HID, projB[t],
                    X + (size_t)OFF[t] * HID, HID, NN[t], HID, 256);

    for (int l = 0; l < 2; ++l) {
        // K/Q/V projections (K=128)
        for (int t = 0; t < 3; ++t) {
            const float* xs = X + (size_t)OFF[t] * HID;
            launch_gemm(xs, HID, L[l].kW[t], HID, L[l].kB[t],
                        Kb + (size_t)OFF[t] * HID, HID, NN[t], HID, HID);
            launch_gemm(xs, HID, L[l].qW[t], HID, L[l].qB[t],
                        Qb + (size_t)OFF[t] * HID, HID, NN[t], HID, HID);
            launch_gemm(xs, HID, L[l].vW[t], HID, L[l].vB[t],
                        Vb + (size_t)OFF[t] * HID, HID, NN[t], HID, HID);
        }

        // init segment buffers
        {
            const size_t nmh = (size_t)NTOT * H_HEADS;
            fill_u32_k<<<(unsigned)((nmh + 255) / 256), blk256, 0, stream>>>(
                SMAX, 0x007FFFFFu /* enc(-inf) */, nmh);
            fill_u32_k<<<(unsigned)((nmh + 255) / 256), blk256, 0, stream>>>(
                (unsigned*)SSUM, 0u, nmh);
            fill_u32_k<<<(unsigned)((F128 + 255) / 256), blk256, 0, stream>>>(
                (unsigned*)AGG, 0u, F128);
        }

        const unsigned ewaves_blocks = (unsigned)(((size_t)EE * 32 + 255) / 256);
        const unsigned eth_blocks    = (unsigned)(((size_t)EE * H_HEADS + 255) / 256);

        // pass 1: logits + segment max
        for (int e = 0; e < 6; ++e)
            edge_logit_k<<<ewaves_blocks, blk256, 0, stream>>>(
                ei[e], EE, Kb, Qb, OFF[ESRC[e]], OFF[EDST[e]],
                L[l].arel[e], L[l].prel[e],
                LOGIT + (size_t)e * EE * H_HEADS, SMAX);

        // pass 2: exp + segment sum
        for (int e = 0; e < 6; ++e)
            edge_exp_k<<<eth_blocks, blk256, 0, stream>>>(
                ei[e], EE, OFF[EDST[e]],
                LOGIT + (size_t)e * EE * H_HEADS, SMAX, SSUM);

        // pass 3: alpha-weighted messages -> AGG
        for (int e = 0; e < 6; ++e)
            edge_msg_k<<<ewaves_blocks, blk256, 0, stream>>>(
                ei[e], EE, Vb, OFF[ESRC[e]], OFF[EDST[e]],
                L[l].mrel[e], LOGIT + (size_t)e * EE * H_HEADS, SSUM, AGG);

        // gelu(AGG) in place, then a_lin: CO = gelu(AGG) @ aW + aB
        gelu_inplace_k<<<(unsigned)((F128 + 255) / 256), blk256, 0, stream>>>(
            AGG, F128);
        for (int t = 0; t < 3; ++t)
            launch_gemm(AGG + (size_t)OFF[t] * HID, HID, L[l].aW[t], HID,
                        L[l].aB[t], CO + (size_t)OFF[t] * HID, HID,
                        NN[t], HID, HID);

        // skip-gate + residual + LayerNorm + GELU (updates X in place)
        for (int t = 0; t < 3; ++t) {
            const unsigned blocks = (unsigned)(((size_t)NN[t] * 32 + 255) / 256);
            ln_update_k<<<blocks, blk256, 0, stream>>>(
                CO, X, L[l].skip[t], L[l].lng[t], L[l].lnb[t], OFF[t], NN[t]);
        }
    }

    // ---- fused decoder over 200000 pairs ----
    decoder_fused<<<NPAIRS / 16, dim3(128), 0, stream>>>(
        X + (size_t)OFF[0] * HID,  // drug features
        X + (size_t)OFF[1] * HID,  // disease features
        drug_idx, dis_idx,
        dW[0], dB[0], dW[1], dB[1], dW[2], dB[2],
        (float*)d_out, NPAIRS);
}